// FNNAttention_13993003451111
// MI455X (gfx1250) — compile-verified
//
#include <hip/hip_runtime.h>
#include <math.h>

typedef __attribute__((ext_vector_type(2))) float v2f;
typedef __attribute__((ext_vector_type(8))) float v8f;

#define B_  4
#define Q_  256
#define M_  256
#define QS_ 512
#define KS_ 512
#define H_  512
#define CC_ (QS_ + KS_)   // 1024
#define TQ_ 8             // query rows register-tiled per score block

// Hardware tanh (V_TANH_F32 on gfx1250). Prefer the clang builtin (correct
// TRANS hazard scheduling); otherwise emit the instruction directly, with a
// trailing v_nop per the ISA TRANS-op hazard rule (1 indep op / V_NOP before
// the result may be consumed).
__device__ __forceinline__ float fast_tanh(float x) {
#if __has_builtin(__builtin_amdgcn_tanhf)
  return __builtin_amdgcn_tanhf(x);
#elif __has_builtin(__builtin_amdgcn_tanh_f32)
  return __builtin_amdgcn_tanh_f32(x);
#else
  float r;
  asm("v_tanh_f32 %0, %1\n\tv_nop" : "=v"(r) : "v"(x));
  return r;
#endif
}

// ---------------------------------------------------------------------------
// Y = X * W^T + bias,  X:[Bn][N][D] f32, W:[HOUT][D] f32.
// One wave32 computes one 16x16 output tile with V_WMMA_F32_16X16X4_F32.
// A frag (16x4): half=lane>>4, r=lane&15 -> a[j] = X[r][k0+2*half+j]
// B frag (4x16): b[j] = (W^T)[k0+2*half+j][n] = W[h0+r][k0+2*half+j]
// D: element v -> (row n0+v+8*half, col h0+r)
// TRANSPOSE=true stores Y as [Bn][HOUT][N] (lane-contiguous score reads).
// ---------------------------------------------------------------------------
template <int N, int D, int HOUT, bool TRANSPOSE>
__global__ void proj_gemm_wmma(const float* __restrict__ X,
                               const float* __restrict__ W,
                               const float* __restrict__ bias,
                               float* __restrict__ Y) {
  const int lane = threadIdx.x & 31;
  const int wave = threadIdx.x >> 5;
  const int half = lane >> 4;
  const int r    = lane & 15;

  const int tilesN = N / 16, tilesH = HOUT / 16;
  int tile = blockIdx.x * (blockDim.x >> 5) + wave;
  const int b  = tile / (tilesN * tilesH);
  int t        = tile % (tilesN * tilesH);
  const int n0 = (t / tilesH) * 16;
  const int h0 = (t % tilesH) * 16;

  const float* Xrow = X + ((size_t)b * N + (n0 + r)) * D;
  const float* Wrow = W + (size_t)(h0 + r) * D;

  v8f acc = {};
  for (int k0 = 0; k0 < D; k0 += 4) {
    __builtin_prefetch(Xrow + k0 + 64, 0, 0);
    v2f a  = *(const v2f*)(Xrow + k0 + 2 * half);
    v2f bb = *(const v2f*)(Wrow + k0 + 2 * half);
    acc = __builtin_amdgcn_wmma_f32_16x16x4_f32(false, a, false, bb,
                                                (short)0, acc, false, false);
  }
  const float bv = bias[h0 + r];
#pragma unroll
  for (int v = 0; v < 8; ++v) {
    const int row = n0 + v + 8 * half;   // index in N
    const int col = h0 + r;              // index in HOUT
    const float val = acc[v] + bv;
    if (TRANSPOSE) Y[((size_t)b * HOUT + col) * N + row] = val;
    else           Y[((size_t)b * N + row) * HOUT + col] = val;
  }
}

// ---------------------------------------------------------------------------
// Scores + masked softmax, register-tiled over TQ_ query rows per block.
// Block = (b, q-tile), 256 threads; thread m owns memory slot m.
// logits[q,m] = sum_h Wl[h]*tanh(qproj[b,q,h] + kT[b,h,m]) + bl; mask -> -1e18.
// kT is read exactly once per block and reused across TQ_ accumulators.
// ---------------------------------------------------------------------------
__global__ void bahdanau_scores(const float* __restrict__ qproj,   // [B][Q][H]
                                const float* __restrict__ kT,      // [B][H][M]
                                const unsigned char* __restrict__ mask, // [B][M]
                                const float* __restrict__ Wl,      // [H]
                                const float* __restrict__ bl,      // [1]
                                float* __restrict__ weights) {     // [B][Q][M]
  __shared__ float qs[TQ_][H_];   // 16 KB
  __shared__ float wl[H_];        //  2 KB
  __shared__ float partial[8];    // per-wave reduction partials
  const int b    = blockIdx.y;
  const int q0   = blockIdx.x * TQ_;
  const int m    = threadIdx.x;        // 0..255
  const int lane = m & 31;
  const int wv   = m >> 5;

  for (int i = threadIdx.x; i < TQ_ * H_; i += blockDim.x)
    qs[i / H_][i % H_] = qproj[((size_t)b * Q_ + q0 + i / H_) * H_ + (i % H_)];
  for (int h = threadIdx.x; h < H_; h += blockDim.x) wl[h] = Wl[h];
  __syncthreads();

  const float* kcol = kT + (size_t)b * H_ * M_ + m;
  float acc[TQ_];
#pragma unroll
  for (int qq = 0; qq < TQ_; ++qq) acc[qq] = 0.f;

  for (int h = 0; h < H_; ++h) {
    const float kv = kcol[(size_t)h * M_];   // coalesced, read once per block
    const float w  = wl[h];
#pragma unroll
    for (int qq = 0; qq < TQ_; ++qq)
      acc[qq] = fmaf(w, fast_tanh(qs[qq][h] + kv), acc[qq]);
  }

  const float blv    = bl[0];
  const bool  masked = mask[b * M_ + m] != 0;

#pragma unroll
  for (int qq = 0; qq < TQ_; ++qq) {
    const float lg = masked ? -1e18f : (acc[qq] + blv);
    // --- block max: wave32 butterfly + 8-partial hop ---
    float mx = lg;
#pragma unroll
    for (int off = 16; off > 0; off >>= 1) mx = fmaxf(mx, __shfl_xor(mx, off));
    if (lane == 0) partial[wv] = mx;
    __syncthreads();
    mx = partial[0];
#pragma unroll
    for (int i = 1; i < 8; ++i) mx = fmaxf(mx, partial[i]);
    __syncthreads();
    // --- block sum ---
    const float e = __expf(lg - mx);
    float s = e;
#pragma unroll
    for (int off = 16; off > 0; off >>= 1) s += __shfl_xor(s, off);
    if (lane == 0) partial[wv] = s;
    __syncthreads();
    s = partial[0];
#pragma unroll
    for (int i = 1; i < 8; ++i) s += partial[i];
    __syncthreads();
    weights[((size_t)b * Q_ + q0 + qq) * M_ + m] = e / s;
  }
}

// ---------------------------------------------------------------------------
// attns[b,q,d] = sum_m weights[b,q,m] * memory[b,m,d]   (WMMA f32)
// ---------------------------------------------------------------------------
__global__ void attn_gemm_wmma(const float* __restrict__ weights, // [B][Q][M]
                               const float* __restrict__ memory,  // [B][M][KS]
                               float* __restrict__ attns) {       // [B][Q][KS]
  const int lane = threadIdx.x & 31;
  const int wave = threadIdx.x >> 5;
  const int half = lane >> 4;
  const int r    = lane & 15;

  const int tilesQ = Q_ / 16, tilesD = KS_ / 16;
  int tile = blockIdx.x * (blockDim.x >> 5) + wave;
  const int b  = tile / (tilesQ * tilesD);
  int t        = tile % (tilesQ * tilesD);
  const int q0 = (t / tilesD) * 16;
  const int d0 = (t % tilesD) * 16;

  const float* Arow  = weights + ((size_t)b * Q_ + (q0 + r)) * M_;
  const float* Bbase = memory + (size_t)b * M_ * KS_;

  v8f acc = {};
  for (int k0 = 0; k0 < M_; k0 += 4) {
    v2f a = *(const v2f*)(Arow + k0 + 2 * half);
    v2f bb;
    bb.x = Bbase[(size_t)(k0 + 2 * half + 0) * KS_ + d0 + r];
    bb.y = Bbase[(size_t)(k0 + 2 * half + 1) * KS_ + d0 + r];
    acc = __builtin_amdgcn_wmma_f32_16x16x4_f32(false, a, false, bb,
                                                (short)0, acc, false, false);
  }
#pragma unroll
  for (int v = 0; v < 8; ++v)
    attns[((size_t)b * Q_ + (q0 + v + 8 * half)) * KS_ + d0 + r] = acc[v];
}

// ---------------------------------------------------------------------------
// out[b,q,o] = tanh( [query|attns][b,q,:] . Wo[o,:] + bo[o] )   (WMMA f32)
// ---------------------------------------------------------------------------
__global__ void out_gemm_wmma(const float* __restrict__ query, // [B][Q][QS]
                              const float* __restrict__ attns, // [B][Q][KS]
                              const float* __restrict__ Wo,    // [QS][CC]
                              const float* __restrict__ bo,    // [QS]
                              float* __restrict__ out) {       // [B][Q][QS]
  const int lane = threadIdx.x & 31;
  const int wave = threadIdx.x >> 5;
  const int half = lane >> 4;
  const int r    = lane & 15;

  const int tilesQ = Q_ / 16, tilesO = QS_ / 16;
  int tile = blockIdx.x * (blockDim.x >> 5) + wave;
  const int b  = tile / (tilesQ * tilesO);
  int t        = tile % (tilesQ * tilesO);
  const int q0 = (t / tilesO) * 16;
  const int o0 = (t % tilesO) * 16;

  const float* Qrow = query + ((size_t)b * Q_ + (q0 + r)) * QS_;
  const float* Arow = attns + ((size_t)b * Q_ + (q0 + r)) * KS_;
  const float* Wrow = Wo + (size_t)(o0 + r) * CC_;

  v8f acc = {};
  for (int k0 = 0; k0 < CC_; k0 += 4) {
    const int kk = k0 + 2 * half;                    // even; never straddles 512
    v2f a  = (kk < QS_) ? *(const v2f*)(Qrow + kk)
                        : *(const v2f*)(Arow + (kk - QS_));
    v2f bb = *(const v2f*)(Wrow + kk);
    acc = __builtin_amdgcn_wmma_f32_16x16x4_f32(false, a, false, bb,
                                                (short)0, acc, false, false);
  }
  const float bv = bo[o0 + r];
#pragma unroll
  for (int v = 0; v < 8; ++v)
    out[((size_t)b * Q_ + (q0 + v + 8 * half)) * QS_ + o0 + r] =
        fast_tanh(acc[v] + bv);
}

// ---------------------------------------------------------------------------
extern "C" void kernel_launch(void* const* d_in, const int* in_sizes, int n_in,
                              void* d_out, int out_size, void* d_ws, size_t ws_size,
                              hipStream_t stream) {
  const float*         query  = (const float*)d_in[0];
  const float*         memory = (const float*)d_in[1];
  const unsigned char* mask   = (const unsigned char*)d_in[2]; // jnp.bool_ (1B)
  const float*         Wk     = (const float*)d_in[3];
  const float*         bk     = (const float*)d_in[4];
  const float*         Wq     = (const float*)d_in[5];
  const float*         bq     = (const float*)d_in[6];
  const float*         Wl     = (const float*)d_in[7];
  const float*         bl     = (const float*)d_in[8];
  const float*         Wo     = (const float*)d_in[9];
  const float*         bo     = (const float*)d_in[10];

  float* out     = (float*)d_out;                       // [B][Q][QS]
  float* weights = out + (size_t)B_ * Q_ * QS_;         // [B][Q][M]

  float* qproj = (float*)d_ws;                          // B*Q*H
  float* kT    = qproj + (size_t)B_ * Q_ * H_;          // B*H*M (transposed)
  float* attns = kT + (size_t)B_ * H_ * M_;             // B*Q*KS

  // q projection: [B,Q,QS] x Wq^T -> [B,Q,H]
  proj_gemm_wmma<Q_, QS_, H_, false>
      <<<(B_ * (Q_ / 16) * (H_ / 16)) / 8, 256, 0, stream>>>(query, Wq, bq, qproj);
  // k projection (stored transposed): [B,M,KS] x Wk^T -> kT [B,H,M]
  proj_gemm_wmma<M_, KS_, H_, true>
      <<<(B_ * (M_ / 16) * (H_ / 16)) / 8, 256, 0, stream>>>(memory, Wk, bk, kT);

  dim3 sgrid(Q_ / TQ_, B_);
  bahdanau_scores<<<sgrid, M_, 0, stream>>>(qproj, kT, mask, Wl, bl, weights);

  attn_gemm_wmma<<<(B_ * (Q_ / 16) * (KS_ / 16)) / 8, 256, 0, stream>>>(
      weights, memory, attns);

  out_gemm_wmma<<<(B_ * (Q_ / 16) * (QS_ / 16)) / 8, 256, 0, stream>>>(
      query, attns, Wo, bo, out);
}